// EnergyForcesModelBase_86337432584822
// MI455X (gfx1250) — compile-verified
//
#include <hip/hip_runtime.h>
#include <hip/hip_bf16.h>

// ---------------------------------------------------------------------------
// EnergyForces on MI455X (gfx1250, wave32).
//
// Bandwidth-bound problem (~430 MB read / call -> ~19 us at 23.3 TB/s HBM).
// Both GEMVs are done with the native fp32 matrix op V_WMMA_F32_16X16X4_F32:
//   A = 16 rows of h (documented 32-bit 16x4 A layout),
//   B = W broadcast across all 16 columns  =>  D(M,N) = dot(row M, W) for all N,
// so each lane holds the 8 dot products of its half's rows and the segment-sums reduce to
// per-lane global_atomic_add_f32 (no cross-lane reduction, no LDS).
//
// Layout assumptions (compile-only environment, cannot be runtime-verified):
//   A 16x4 f32 (ISA table): lane L -> row L%16; v0 = K+ (L<16?0:2), v1 = K+ (L<16?1:3)
//   B 4x16 f32 (assumed mirror, half-major): v0 = W[K + (L<16?0:2)], v1 = W[K + (L<16?1:3)]
//   D 16x16 f32 (ISA table): lane L, VGPR v -> M = v + 8*(L>=16), N = L%16
// ---------------------------------------------------------------------------

typedef __attribute__((ext_vector_type(2))) float v2f;
typedef __attribute__((ext_vector_type(8))) float v8f;

#define N_ATOMS   100000
#define N_EDGES   625000
#define N_MOL     1000
#define EMB_ATOM  256
#define EMB_EDGE  128

// select acc[s] for s in [0,8) without scratch spills (cndmask chain)
__device__ __forceinline__ float sel8(v8f a, int s) {
    float v = a[0];
    v = (s == 1) ? a[1] : v;
    v = (s == 2) ? a[2] : v;
    v = (s == 3) ? a[3] : v;
    v = (s == 4) ? a[4] : v;
    v = (s == 5) ? a[5] : v;
    v = (s == 6) ? a[6] : v;
    v = (s == 7) ? a[7] : v;
    return v;
}

// ---------------------------------------------------------------------------
// Kernel 1: e_atom = h_energy @ W_energy ; y[mol] += e_atom  (batch_idx sorted)
// One wave handles a 16-atom tile; K = 256 -> 64 wmma steps.
// ---------------------------------------------------------------------------
__global__ __launch_bounds__(256) void
ef_energy_kernel(const float* __restrict__ h_energy,
                 const float* __restrict__ W_energy,
                 const int*   __restrict__ batch_idx,
                 float*       __restrict__ y)
{
    const int wave = (blockIdx.x * blockDim.x + threadIdx.x) >> 5;
    const int lane = threadIdx.x & 31;
    const int rowbase = wave * 16;
    if (rowbase >= N_ATOMS) return;          // wave-uniform: EXEC stays all-1s

    const int m       = lane & 15;
    const int halfsel = lane >> 4;           // 0: K+{0,1}, 1: K+{2,3}
    const int kshift  = halfsel ? 2 : 0;

    const float* __restrict__ rowptr = h_energy + (size_t)(rowbase + m) * EMB_ATOM;

    v8f acc = {};
#pragma unroll 4
    for (int kb = 0; kb < EMB_ATOM; kb += 4) {
        const int ko = kb + kshift;
        // streamed exactly once -> non-temporal (h_energy+h_forces > 192MB L2)
        v2f a = __builtin_nontemporal_load((const v2f*)(rowptr + ko));
        v2f b = *(const v2f*)(W_energy + ko);   // hot in WGP$/L2
        acc = __builtin_amdgcn_wmma_f32_16x16x4_f32(
            /*neg_a=*/false, a, /*neg_b=*/false, b,
            /*c_mod=*/(short)0, acc, /*reuse_a=*/false, /*reuse_b=*/false);
    }

    // D: lane half h, VGPR v holds dot(row v + 8h). Lanes 0-7 own rows 0-7,
    // lanes 16-23 own rows 8-15; one atom per active lane.
    const int  s      = lane & 7;
    const int  atom   = rowbase + s + 8 * halfsel;
    const bool active = ((lane & 15) < 8);
    if (active) {
        const float dotv = sel8(acc, s);
        const int   mol  = batch_idx[atom];
        atomicAdd(&y[mol], dotv);            // global_atomic_add_f32, no return
    }
}

// ---------------------------------------------------------------------------
// Kernel 2: f = (h_forces @ W_forces + b) * V_st ; force[idx_t[e]] += f
// One wave handles a 16-edge tile; K = 128 -> 32 wmma steps. Tail predicated
// on the scatter only (loads clamped so EXEC is all-1s for every WMMA).
// ---------------------------------------------------------------------------
__global__ __launch_bounds__(256) void
ef_forces_kernel(const float* __restrict__ h_forces,
                 const float* __restrict__ W_forces,
                 const float* __restrict__ b_forces,
                 const float* __restrict__ V_st,
                 const int*   __restrict__ idx_t,
                 float*       __restrict__ force)
{
    const int wave = (blockIdx.x * blockDim.x + threadIdx.x) >> 5;
    const int lane = threadIdx.x & 31;
    const int rowbase = wave * 16;
    if (rowbase >= N_EDGES) return;          // wave-uniform

    const int m       = lane & 15;
    const int halfsel = lane >> 4;
    const int kshift  = halfsel ? 2 : 0;

    int rowm = rowbase + m;
    if (rowm >= N_EDGES) rowm = N_EDGES - 1; // clamp loads, keep EXEC all-1s
    const float* __restrict__ rowptr = h_forces + (size_t)rowm * EMB_EDGE;

    const float bconst = b_forces[0];        // uniform -> scalar load

    v8f acc = {};
#pragma unroll 4
    for (int kb = 0; kb < EMB_EDGE; kb += 4) {
        const int ko = kb + kshift;
        v2f a = __builtin_nontemporal_load((const v2f*)(rowptr + ko));
        v2f b = *(const v2f*)(W_forces + ko);
        acc = __builtin_amdgcn_wmma_f32_16x16x4_f32(
            false, a, false, b, (short)0, acc, false, false);
    }

    const int  s      = lane & 7;
    const int  e      = rowbase + s + 8 * halfsel;
    const bool active = ((lane & 15) < 8) && (e < N_EDGES);
    if (active) {
        const float f  = sel8(acc, s) + bconst;
        const int   t  = idx_t[e];
        const float v0 = V_st[(size_t)e * 3 + 0];
        const float v1 = V_st[(size_t)e * 3 + 1];
        const float v2 = V_st[(size_t)e * 3 + 2];
        atomicAdd(&force[(size_t)t * 3 + 0], f * v0);
        atomicAdd(&force[(size_t)t * 3 + 1], f * v1);
        atomicAdd(&force[(size_t)t * 3 + 2], f * v2);
    }
}

// ---------------------------------------------------------------------------
// Launch
// ---------------------------------------------------------------------------
extern "C" void kernel_launch(void* const* d_in, const int* in_sizes, int n_in,
                              void* d_out, int out_size, void* d_ws, size_t ws_size,
                              hipStream_t stream) {
    const float* h_energy  = (const float*)d_in[0];
    const float* h_forces  = (const float*)d_in[1];
    const float* V_st      = (const float*)d_in[2];
    const int*   idx_t     = (const int*)  d_in[3];
    const int*   batch_idx = (const int*)  d_in[4];
    const float* W_energy  = (const float*)d_in[5];
    const float* W_forces  = (const float*)d_in[6];
    const float* b_forces  = (const float*)d_in[7];

    float* y     = (float*)d_out;            // N_MOL
    float* force = y + N_MOL;                // N_ATOMS*3, row-major

    // outputs are accumulated via atomics -> zero them every call
    hipMemsetAsync(d_out, 0, (size_t)(N_MOL + N_ATOMS * 3) * sizeof(float), stream);

    // energy: 100000/16 = 6250 tiles, 8 waves (256 thr) per block
    {
        const int tiles  = N_ATOMS / 16;                 // 6250 exactly
        const int blocks = (tiles + 7) / 8;
        ef_energy_kernel<<<blocks, 256, 0, stream>>>(h_energy, W_energy, batch_idx, y);
    }
    // forces: ceil(625000/16) = 39063 tiles
    {
        const int tiles  = (N_EDGES + 15) / 16;
        const int blocks = (tiles + 7) / 8;
        ef_forces_kernel<<<blocks, 256, 0, stream>>>(h_forces, W_forces, b_forces,
                                                     V_st, idx_t, force);
    }
}